// URNN_13709535609650
// MI455X (gfx1250) — compile-verified
//
#include <hip/hip_runtime.h>
#include <math.h>

#define B_    64
#define T_    512
#define IN_   256
#define OUT_  256
#define H_    512
#define BT_   (B_ * T_)
#define TWOH_ (2 * H_)

typedef float v2f __attribute__((ext_vector_type(2)));
typedef float v8f __attribute__((ext_vector_type(8)));

// ---------------------------------------------------------------------------
// WMMA fp32 GEMM:  C[m,n] = sum_k A[m,k] * W[n,k]  (+ bias[n])
// A: M x K row-major, W: N x K row-major (B = W^T), C: M x N row-major.
// Block = 256 threads = 8 waves. Each block owns one 16-row A tile (a
// CONTIGUOUS 16*K*4-byte chunk since tiles span full K) and 8 adjacent
// 16-col N tiles (one per wave). The A tile is staged into LDS once with
// gfx1250 async copy (global_load_async_to_lds_b128, ASYNCcnt) and shared
// by all 8 waves; weights stream from L2 (1 MB, resident in 192 MB L2).
// Fragments per V_WMMA_F32_16X16X4_F32:
//   A (16x4): lanes 0-15 -> M=lane, K={0,1}; lanes 16-31 -> M=lane-16, K={2,3}
//   B (4x16): symmetric (N=lane&15, K pair by lane>=16)
//   C/D (16x16): VGPR r -> M=r (lanes 0-15) / M=r+8 (lanes 16-31), N=lane&15
// ---------------------------------------------------------------------------
template <int K>
__global__ void __launch_bounds__(256) wmma_gemm_nt(const float* __restrict__ A,
                                                    const float* __restrict__ W,
                                                    const float* __restrict__ bias,
                                                    float* __restrict__ C,
                                                    int N) {
  __shared__ __align__(16) float sA[16 * K];

  const int tid    = (int)threadIdx.x;
  const int lane   = tid & 31;
  const int wv     = tid >> 5;             // wave in block: 0..7
  const int nBlk   = N >> 7;               // N / 128
  const int tm     = (int)blockIdx.x / nBlk;
  const int tnBlk  = (int)blockIdx.x - tm * nBlk;
  const int tn     = tnBlk * 8 + wv;

  // --- stage the 16 x K A tile into LDS via async copy (B128 per lane) ---
  {
    const unsigned bytes   = 16u * K * 4u;
    const unsigned ldsBase = (unsigned)(size_t)&sA[0];
    const unsigned long long src = (unsigned long long)(A + (size_t)tm * 16 * K);
    for (unsigned off = (unsigned)tid * 16u; off < bytes; off += 256u * 16u) {
      unsigned lds = ldsBase + off;
      asm volatile("global_load_async_to_lds_b128 %0, %1, %2 offset:0"
                   :: "v"(lds), "v"(off), "s"(src) : "memory");
    }
    asm volatile("s_wait_asynccnt 0" ::: "memory");
    __syncthreads();
  }

  const int r  = lane & 15;                // row in A tile / col in B tile
  const int kb = (lane >> 4) << 1;         // 0 (lanes 0-15) or 2 (lanes 16-31)

  const float* aRow = &sA[r * K];
  const float* Wrow = W + (size_t)(tn * 16 + r) * K;

  v8f acc = {0.f, 0.f, 0.f, 0.f, 0.f, 0.f, 0.f, 0.f};
#pragma unroll 8
  for (int k = 0; k < K; k += 4) {
    v2f a, b;
    a.x = aRow[k + kb];
    a.y = aRow[k + kb + 1];
    b.x = Wrow[k + kb];
    b.y = Wrow[k + kb + 1];
    acc = __builtin_amdgcn_wmma_f32_16x16x4_f32(false, a, false, b,
                                                (short)0, acc, false, false);
  }

  const int col  = tn * 16 + r;
  const int row0 = tm * 16 + ((lane >> 4) << 3);
  const float bv = bias ? bias[col] : 0.f;
#pragma unroll
  for (int i = 0; i < 8; ++i)
    C[(size_t)(row0 + i) * N + col] = acc[i] + bv;
}

// ---------------------------------------------------------------------------
// Block-wide complex sum over 512 threads. Single internal barrier; callers
// guarantee >=1 __syncthreads() between the last read of rA/rB and re-entry
// (always true here: >=9 FFT barriers separate consecutive calls).
// ---------------------------------------------------------------------------
__device__ __forceinline__ void blockReduce2(float& x, float& y,
                                             float* rA, float* rB, int tid) {
#pragma unroll
  for (int off = 16; off > 0; off >>= 1) {
    x += __shfl_down(x, off, 32);
    y += __shfl_down(y, off, 32);
  }
  if ((tid & 31) == 0) { rA[tid >> 5] = x; rB[tid >> 5] = y; }
  __syncthreads();
  float sx = 0.f, sy = 0.f;
#pragma unroll
  for (int w = 0; w < 16; ++w) { sx += rA[w]; sy += rB[w]; }
  x = sx; y = sy;
}

// ---------------------------------------------------------------------------
// Sequential unitary-RNN scan. One block per batch row; 512 threads, one
// complex element per thread. Forward FFT = radix-2 DIF (natural -> bitrev),
// inverse = radix-2 DIT (bitrev -> natural); frequency-domain elementwise
// constants are pre-indexed in bit-reversed space, so no reorder pass.
// LDS exchange uses ping-pong buffers: ONE barrier per stage (a write to
// buffer p at use n+2 is fenced by the barrier of use n+1, which postdates
// all reads at use n). 21+1 barriers per timestep vs 40 for naive.
// UO rows are read (u) then overwritten (output) in place.
// ---------------------------------------------------------------------------
__global__ void __launch_bounds__(512) urnn_scan(
    const float* __restrict__ h0, const float* __restrict__ b_h,
    const float* __restrict__ d1, const float* __restrict__ d2,
    const float* __restrict__ d3,
    const float* __restrict__ r1re, const float* __restrict__ r1im,
    const float* __restrict__ r2re, const float* __restrict__ r2im,
    const int* __restrict__ perm,
    float* __restrict__ UO,         // (BT, 1024): in = u (re|im), out = outs
    float* __restrict__ hidden) {   // (B, 1024) final hidden state
  __shared__ float twc[256], tws[256];
  __shared__ float bR[2][512], bI[2][512];
  __shared__ float redA[16], redB[16];

  const int tid = (int)threadIdx.x;
  const int b   = (int)blockIdx.x;
  const int br  = (int)(__brev((unsigned)tid) >> 23);   // 9-bit reversal

  if (tid < 256) {
    float th = 6.283185307179586f * (float)tid * (1.0f / 512.0f);
    twc[tid] = cosf(th);
    tws[tid] = sinf(th);
  }

  float a1 = d1[tid];  float e1c = cosf(a1), e1s = sinf(a1);
  float a2 = d2[br];   float e2c = cosf(a2), e2s = sinf(a2);
  float a3 = d3[tid];
  // fold the IFFT 1/512 normalization into e3 (reflection 2 is linear)
  float e3c = cosf(a3) * (1.0f / 512.0f), e3s = sinf(a3) * (1.0f / 512.0f);
  float v1r = r1re[br],  v1i = r1im[br];
  float v2r = r2re[tid], v2i = r2im[tid];
  float bh  = b_h[tid];
  int   pBR = (int)(__brev((unsigned)perm[br]) >> 23);

  float n1 = v1r * v1r + v1i * v1i;
  float n2 = v2r * v2r + v2i * v2i;
  __syncthreads();                          // twiddle init + redA/B clean
  blockReduce2(n1, n2, redA, redB, tid);
  const float fac1 = 2.0f / n1;
  const float fac2 = 2.0f / n2;

  float hr = h0[(size_t)b * TWOH_ + tid];
  float hi = h0[(size_t)b * TWOH_ + H_ + tid];

  for (int t = 0; t < T_; ++t) {
    __syncthreads();                        // fence last step's buffer reads
    float* row = UO + ((size_t)b * T_ + t) * TWOH_;
    float ur = row[tid];
    float ui = row[H_ + tid];

    // s = h * e1
    float sr = hr * e1c - hi * e1s;
    float si = hr * e1s + hi * e1c;

    int ph = 0;
    // forward FFT-512, radix-2 DIF (output bit-reversed), 1 barrier/stage
#pragma unroll
    for (int st = 8; st >= 0; --st) {
      int span = 1 << st;
      bR[ph][tid] = sr; bI[ph][tid] = si;
      __syncthreads();
      float pr = bR[ph][tid ^ span], pi = bI[ph][tid ^ span];
      ph ^= 1;
      if ((tid & span) == 0) {
        sr += pr; si += pi;
      } else {
        float ar = pr - sr, ai = pi - si;
        int tw = (tid & (span - 1)) << (8 - st);
        float c = twc[tw], s = tws[tw];     // w = c - i*s
        sr = ar * c + ai * s;
        si = ai * c - ar * s;
      }
    }

    // reflection 1 (bitrev-indexed v1):  s -= fac1 * (s . conj(v1)) * v1
    float dr = sr * v1r + si * v1i;
    float di = si * v1r - sr * v1i;
    blockReduce2(dr, di, redA, redB, tid);
    sr -= fac1 * (dr * v1r - di * v1i);
    si -= fac1 * (dr * v1i + di * v1r);

    // permutation (bitrev-remapped) then * e2
    bR[ph][tid] = sr; bI[ph][tid] = si;
    __syncthreads();
    float qr = bR[ph][pBR], qi = bI[ph][pBR];
    ph ^= 1;
    sr = qr * e2c - qi * e2s;
    si = qr * e2s + qi * e2c;

    // inverse FFT-512, radix-2 DIT (bitrev input -> natural output)
#pragma unroll
    for (int st = 0; st <= 8; ++st) {
      int span = 1 << st;
      bR[ph][tid] = sr; bI[ph][tid] = si;
      __syncthreads();
      float pr = bR[ph][tid ^ span], pi = bI[ph][tid ^ span];
      ph ^= 1;
      int tw = (tid & (span - 1)) << (8 - st);
      float c = twc[tw], s = tws[tw];       // inverse w = c + i*s
      if ((tid & span) == 0) {
        float bwr = pr * c - pi * s, bwi = pi * c + pr * s;
        sr += bwr; si += bwi;
      } else {
        float bwr = sr * c - si * s, bwi = si * c + sr * s;
        sr = pr - bwr; si = pi - bwi;
      }
    }

    // reflection 2 (natural order; 1/512 deferred to e3)
    dr = sr * v2r + si * v2i;
    di = si * v2r - sr * v2i;
    blockReduce2(dr, di, redA, redB, tid);
    sr -= fac2 * (dr * v2r - di * v2i);
    si -= fac2 * (dr * v2i + di * v2r);

    // pre = u + s*e3 ; modReLU
    float prr = ur + (sr * e3c - si * e3s);
    float pri = ui + (sr * e3s + si * e3c);
    float nrm = sqrtf(prr * prr + pri * pri);
    float sc  = fmaxf(nrm + bh, 0.0f) / (nrm + 1e-6f);
    hr = prr * sc;
    hi = pri * sc;

    row[tid]      = hr;   // overwrite u row with output row
    row[H_ + tid] = hi;
  }

  hidden[(size_t)b * TWOH_ + tid]      = hr;
  hidden[(size_t)b * TWOH_ + H_ + tid] = hi;
}

// ---------------------------------------------------------------------------
extern "C" void kernel_launch(void* const* d_in, const int* in_sizes, int n_in,
                              void* d_out, int out_size, void* d_ws, size_t ws_size,
                              hipStream_t stream) {
  const float* x      = (const float*)d_in[0];
  const float* h0     = (const float*)d_in[1];
  const float* itoh_w = (const float*)d_in[2];
  const float* fc_w   = (const float*)d_in[3];
  const float* fc_b   = (const float*)d_in[4];
  const float* b_h    = (const float*)d_in[5];
  const float* d1     = (const float*)d_in[6];
  const float* d2     = (const float*)d_in[7];
  const float* d3     = (const float*)d_in[8];
  const float* r1re   = (const float*)d_in[9];
  const float* r1im   = (const float*)d_in[10];
  const float* r2re   = (const float*)d_in[11];
  const float* r2im   = (const float*)d_in[12];
  const int*   perm   = (const int*)d_in[13];

  float* out = (float*)d_out;                 // outputs (B,T,OUT) then hidden (B,2H)
  float* UO  = (float*)d_ws;                  // (BT, 2H) fp32 = 128 MB scratch

  // 1) xp = x @ itoh_w^T -> UO   (M=32768, N=1024, K=256)
  {
    int blocks = (BT_ / 16) * (TWOH_ / 128);  // 2048 * 8 = 16384
    wmma_gemm_nt<IN_><<<blocks, 256, 0, stream>>>(x, itoh_w, nullptr, UO, TWOH_);
  }

  // 2) sequential scan: UO (u) -> UO (outs); final hidden appended to d_out
  urnn_scan<<<B_, 512, 0, stream>>>(h0, b_h, d1, d2, d3,
                                    r1re, r1im, r2re, r2im, perm,
                                    UO, out + (size_t)BT_ * OUT_);

  // 3) outputs = outs @ fc_w^T + fc_b   (M=32768, N=256, K=1024)
  {
    int blocks = (BT_ / 16) * (OUT_ / 128);   // 2048 * 2 = 4096
    wmma_gemm_nt<TWOH_><<<blocks, 256, 0, stream>>>(UO, fc_w, fc_b, out, OUT_);
  }
}